// GraphConvNet_7378753814809
// MI455X (gfx1250) — compile-verified
//
#include <hip/hip_runtime.h>

// ---------------------------------------------------------------------------
// GraphConvNet on MI455X (gfx1250): f32 WMMA pipeline + TDM staging
//   prep1: S_i = A_i @ A_i  (exact f32), pack A0/A1/A2 transposed+padded
//   prep2: pack squares transposed+padded
//   per chunk of CN batch items:
//     K1 (nconv):  h[k][c] = M_k^T @ X_c  for M = {A0,A0^2,A1,A1^2,A2,A2^2}
//                  X staged global->LDS via TENSOR_LOAD_TO_LDS (TDM); jobs
//                  paired over (k,k+1) so each LDS B operand feeds 2 WMMAs
//                  (halves LDS traffic -> below the 256 B/clk WGP roofline)
//     K2 (mix)  :  out = W @ [x ; h]  + b     (K = 224 GEMM)
// All matmuls via v_wmma_f32_16x16x4_f32 (exact f32 vs the f32 reference).
// ---------------------------------------------------------------------------

typedef __attribute__((ext_vector_type(2))) float v2f;
typedef __attribute__((ext_vector_type(8))) float v8f;
typedef __attribute__((ext_vector_type(4))) unsigned int v4u;
typedef __attribute__((ext_vector_type(4))) int v4i;
typedef __attribute__((ext_vector_type(8))) int v8i;

#ifndef __has_builtin
#define __has_builtin(x) 0
#endif
#if __has_builtin(__builtin_amdgcn_tensor_load_to_lds)
#define HAVE_TDM 1
#else
#define HAVE_TDM 0
#endif

#define V 207
#define L 128
#define NB 32
#define NC 32
#define CO 32
#define NPOS (V * L)           // 26496 positions per (n,c) plane
#define NTILES (NPOS / 16)     // 1656, exact
#define KP 208                 // padded K (v) dim
#define LP 144                 // LDS row pitch in floats (2*144 % 64 == 32)
#define ATP_FLOATS (6 * KP * KP)        // 259584
#define HB_OFF 388144                   // 16B-aligned float offset for h buffer

static __device__ __forceinline__ v8f wmma_f32(v2f a, v2f b, v8f c) {
  return __builtin_amdgcn_wmma_f32_16x16x4_f32(false, a, false, b, (short)0, c,
                                               false, false);
}

// ---------------------------------------------------------------------------
// prep1: squares (f32 FMA, tiny) + pack AT for A0,A1,A2 (slots 0,2,4)
// ATP[slot][w][v] = M[v][w], zero-padded to 208x208 (K-pad col/row = 0)
// ---------------------------------------------------------------------------
__global__ void prep1(const float* __restrict__ A0, const float* __restrict__ A1,
                      const float* __restrict__ A2, float* __restrict__ SQ,
                      float* __restrict__ ATP) {
  int idx = blockIdx.x * 256 + threadIdx.x;
  if (idx < 3 * V * V) {
    int i = idx / (V * V);
    int rem = idx - i * (V * V);
    int u = rem / V, w = rem - u * V;
    const float* A = (i == 0) ? A0 : (i == 1) ? A1 : A2;
    float s = 0.f;
    for (int v = 0; v < V; ++v) s = fmaf(A[u * V + v], A[v * V + w], s);
    SQ[idx] = s;  // S[i][u][w] = (A_i^2)[u][w]
    return;
  }
  idx -= 3 * V * V;
  if (idx < 3 * KP * KP) {
    int i = idx / (KP * KP);
    int rem = idx - i * (KP * KP);
    int w = rem / KP, v = rem - w * KP;
    const float* A = (i == 0) ? A0 : (i == 1) ? A1 : A2;
    float val = (w < V && v < V) ? A[v * V + w] : 0.f;
    ATP[(size_t)(2 * i) * (KP * KP) + w * KP + v] = val;
  }
}

// prep2: pack squares into slots 1,3,5
__global__ void prep2(const float* __restrict__ SQ, float* __restrict__ ATP) {
  int idx = blockIdx.x * 256 + threadIdx.x;
  if (idx >= 3 * KP * KP) return;
  int i = idx / (KP * KP);
  int rem = idx - i * (KP * KP);
  int w = rem / KP, v = rem - w * KP;
  float val = (w < V && v < V) ? SQ[(size_t)i * (V * V) + v * V + w] : 0.f;
  ATP[(size_t)(2 * i + 1) * (KP * KP) + w * KP + v] = val;
}

// ---------------------------------------------------------------------------
// K1: nconv. One workgroup per (n,c). 512 threads = 16 waves.
// X[v][l] in LDS, plain row layout with 144-float pitch (TDM hardware pad).
// 39 paired jobs (3 k-pairs x 13 w-tiles). Per job: K-steps outer, 8 N-tiles
// inner; two A panels (slabs k,k+1) share every B operand -> 16 WMMAs per
// K-step at half the LDS traffic. 128 accumulator VGPRs live; no spills.
// ---------------------------------------------------------------------------
__global__ __launch_bounds__(512) void nconv_k(const float* __restrict__ x,
                                               const float* __restrict__ atp,
                                               float* __restrict__ hbuf, int n0) {
  extern __shared__ float xls[];          // KP*LP floats = 119808 bytes
  const int wg = blockIdx.x;
  const int nl = wg >> 5;                 // local batch index within chunk
  const int c = wg & 31;
  const int tid = threadIdx.x;
  const float* xs = x + ((size_t)((n0 + nl) * NC + c)) * NPOS;

  // zero the K-pad row (v = 207); TDM never touches it
  for (int l = tid; l < L; l += 512) xls[207 * LP + l] = 0.f;

#if HAVE_TDM
  if (tid < 32) {  // one wave issues the DMA; TENSORcnt is per-wave
    const unsigned lds_addr = (unsigned)(size_t)xls;       // LDS byte offset
    const unsigned long long ga = (unsigned long long)(size_t)xs;
    // D# group0: count=1 (valid), lds_addr, 57b global addr, type=2 (image)
    const v4u g0 = {1u, lds_addr, (unsigned)ga,
                    (unsigned)((ga >> 32) & 0x01ffffffu) | (2u << 30)};
    // D# group1: data_size=4B(2), pad_enable, pad_interval=6 (every 128
    // dwords), pad_amount=15 (16 dwords) -> LDS pitch 144 floats;
    // tensor_dim0=128, tensor_dim1=207, tile 128x207, dim0_stride=128
    const v8i g1 = {(int)((2u << 16) | (1u << 20) | (6u << 22) | (15u << 25)),
                    (int)(128u << 16),  // tensor_dim0[15:0] at bit 48
                    (int)(207u << 16),  // tensor_dim1[15:0] at bit 80
                    (int)(128u << 16),  // tile_dim0 at bit 112
                    207,                // tile_dim1 at bit 128
                    128,                // tensor_dim0_stride[31:0] at bit 160
                    0, 0};
    const v4i zz = {0, 0, 0, 0};
#if __clang_major__ >= 23
    const v8i z8 = {0, 0, 0, 0, 0, 0, 0, 0};
    __builtin_amdgcn_tensor_load_to_lds(g0, g1, zz, zz, z8, 0);
#else
    __builtin_amdgcn_tensor_load_to_lds(g0, g1, zz, zz, 0);
#endif
    __builtin_amdgcn_s_wait_tensorcnt(0);
  }
#else
  for (int i = tid; i < V * 32; i += 512) {
    int v = i >> 5, f4 = i & 31;
    const float4 d = *((const float4*)(xs + (size_t)v * L) + f4);
    float* dst = xls + (size_t)v * LP + f4 * 4;
    dst[0] = d.x; dst[1] = d.y; dst[2] = d.z; dst[3] = d.w;
  }
#endif
  __syncthreads();

  const int wave = tid >> 5;
  const int lane = tid & 31;
  const int m = lane & 15;                // row-in-tile / N-column
  const int h = lane >> 4;                // K-half select

#pragma unroll 1
  for (int p = wave; p < 39; p += 16) {
    const int kp = p / 13;                // k-pair: slabs 2kp, 2kp+1
    const int wt = p - kp * 13;           // which 16-row output tile
    const int k0 = 2 * kp;
    // A-operand panels: row (wt*16+m), cols 4s + 2h + {0,1}
    const float* at0 = atp + (size_t)k0 * (KP * KP) + (size_t)(wt * 16 + m) * KP + 2 * h;
    const float* at1 = at0 + (size_t)(KP * KP);

    v8f acc0[8], acc1[8];
#pragma unroll
    for (int nt = 0; nt < 8; ++nt) {
      acc0[nt] = (v8f){0.f, 0.f, 0.f, 0.f, 0.f, 0.f, 0.f, 0.f};
      acc1[nt] = (v8f){0.f, 0.f, 0.f, 0.f, 0.f, 0.f, 0.f, 0.f};
    }

#pragma unroll 2
    for (int s = 0; s < 52; ++s) {
      const v2f a0 = *(const v2f*)(at0 + 4 * s);      // global b64, reused 8x
      const v2f a1 = *(const v2f*)(at1 + 4 * s);      // global b64, reused 8x
      const float* br = xls + (size_t)(4 * s + 2 * h) * LP + m;
#pragma unroll
      for (int nt = 0; nt < 8; ++nt) {
        v2f b;
        b.x = br[nt * 16];                // rows 4s+2h / 4s+2h+1
        b.y = br[nt * 16 + LP];           // conflict-free banks (2*LP%64==32)
        acc0[nt] = wmma_f32(a0, b, acc0[nt]);         // one B feeds two WMMAs
        acc1[nt] = wmma_f32(a1, b, acc1[nt]);
      }
    }

    float* hdst0 = hbuf + (((size_t)nl * 6 + k0) * NC + c) * NPOS;
    float* hdst1 = hdst0 + (size_t)NC * NPOS;
    const int wbase = wt * 16 + 8 * h;
    if (wt < 12) {                        // wave-uniform: no masking needed
#pragma unroll
      for (int nt = 0; nt < 8; ++nt) {
        const int lcol = nt * 16 + m;
#pragma unroll
        for (int r = 0; r < 8; ++r) {
          hdst0[(size_t)(wbase + r) * L + lcol] = acc0[nt][r];
          hdst1[(size_t)(wbase + r) * L + lcol] = acc1[nt][r];
        }
      }
    } else {                              // last tile: guard w == 207
#pragma unroll
      for (int nt = 0; nt < 8; ++nt) {
        const int lcol = nt * 16 + m;
#pragma unroll
        for (int r = 0; r < 8; ++r) {
          const int w = wbase + r;
          if (w < V) {
            hdst0[(size_t)w * L + lcol] = acc0[nt][r];
            hdst1[(size_t)w * L + lcol] = acc1[nt][r];
          }
        }
      }
    }
  }
}

// ---------------------------------------------------------------------------
// K2: channel mix. out[n,o,pos] = sum_{kc<224} W[o,kc]*h_all[kc,pos] + b[o]
// h_all rows: kc<32 -> x[n,kc]; else hbuf[nl][kc-32]. K=224 = 56 k-steps.
// 512 threads = 16 waves; wave owns 4 N-tiles x 2 M-tiles of accumulators.
// ---------------------------------------------------------------------------
__global__ __launch_bounds__(512) void mix_k(const float* __restrict__ x,
                                             const float* __restrict__ hbuf,
                                             const float* __restrict__ Wm,
                                             const float* __restrict__ bias,
                                             float* __restrict__ out, int n0) {
  const int nl = blockIdx.x / 26;
  const int tb = blockIdx.x - nl * 26;
  const int n = n0 + nl;
  const int wave = threadIdx.x >> 5;
  const int lane = threadIdx.x & 31;
  const int m = lane & 15;
  const int h = lane >> 4;
  const int nt0 = tb * 64 + wave * 4;

  const float* xbase = x + (size_t)n * NC * NPOS;
  const float* hbase = hbuf + (size_t)nl * 6 * NC * NPOS;

  v8f acc[2][4];
#pragma unroll
  for (int mt = 0; mt < 2; ++mt)
#pragma unroll
    for (int j = 0; j < 4; ++j)
      acc[mt][j] = (v8f){0.f, 0.f, 0.f, 0.f, 0.f, 0.f, 0.f, 0.f};

#pragma unroll 2
  for (int s = 0; s < 56; ++s) {
    const int kc0 = 4 * s + 2 * h;        // this lane-half's K rows kc0, kc0+1
    v2f a0 = *(const v2f*)(Wm + (size_t)(0 + m) * 224 + kc0);
    v2f a1 = *(const v2f*)(Wm + (size_t)(16 + m) * 224 + kc0);
    const float* r0 = (kc0 < NC) ? (xbase + (size_t)kc0 * NPOS)
                                 : (hbase + (size_t)(kc0 - NC) * NPOS);
    const float* r1 = (kc0 + 1 < NC) ? (xbase + (size_t)(kc0 + 1) * NPOS)
                                     : (hbase + (size_t)(kc0 + 1 - NC) * NPOS);
#pragma unroll
    for (int j = 0; j < 4; ++j) {
      const int ntile = nt0 + j;
      if (ntile < NTILES) {               // wave-uniform guard (EXEC stays full)
        const int pos = ntile * 16 + m;
        v2f b;
        b.x = r0[pos];
        b.y = r1[pos];
        acc[0][j] = wmma_f32(a0, b, acc[0][j]);
        acc[1][j] = wmma_f32(a1, b, acc[1][j]);
      }
    }
  }

  float* obase = out + (size_t)n * CO * NPOS;
#pragma unroll
  for (int mt = 0; mt < 2; ++mt) {
#pragma unroll
    for (int j = 0; j < 4; ++j) {
      const int ntile = nt0 + j;
      if (ntile < NTILES) {
        const int pos = ntile * 16 + m;
#pragma unroll
        for (int r = 0; r < 8; ++r) {
          const int o = mt * 16 + 8 * h + r;
          obase[(size_t)o * NPOS + pos] = acc[mt][j][r] + bias[o];
        }
      }
    }
  }
}

// ---------------------------------------------------------------------------
extern "C" void kernel_launch(void* const* d_in, const int* in_sizes, int n_in,
                              void* d_out, int out_size, void* d_ws, size_t ws_size,
                              hipStream_t stream) {
  const float* x  = (const float*)d_in[0];
  const float* A0 = (const float*)d_in[1];
  const float* A1 = (const float*)d_in[2];
  const float* A2 = (const float*)d_in[3];
  const float* Wm = (const float*)d_in[4];
  const float* bv = (const float*)d_in[5];
  float* out = (float*)d_out;
  float* ws = (float*)d_ws;

  float* ATP = ws;
  float* SQ  = ws + ATP_FLOATS;
  float* HB  = ws + HB_OFF;

  // Pick the largest chunk (batch items per pipeline stage) fitting workspace.
  int cn = 8;
  while (cn > 1) {
    size_t need = ((size_t)HB_OFF + (size_t)6 * cn * NC * NPOS) * sizeof(float);
    if (need <= ws_size) break;
    cn >>= 1;
  }

  {
    int total = 3 * V * V + 3 * KP * KP;
    prep1<<<(total + 255) / 256, 256, 0, stream>>>(A0, A1, A2, SQ, ATP);
  }
  {
    int total = 3 * KP * KP;
    prep2<<<(total + 255) / 256, 256, 0, stream>>>(SQ, ATP);
  }

  const int smem = KP * LP * (int)sizeof(float);  // 119808 B (< 320 KB/WGP)
  for (int n0 = 0; n0 < NB; n0 += cn) {
    nconv_k<<<dim3(32 * cn), dim3(512), smem, stream>>>(x, ATP, HB, n0);
    mix_k<<<dim3(26 * cn), dim3(512), 0, stream>>>(x, HB, Wm, bv, out, n0);
  }
}